// EvidenceLevelAttention_21139829031444
// MI455X (gfx1250) — compile-verified
//
#include <hip/hip_runtime.h>
#include <hip/hip_bf16.h>

typedef __attribute__((ext_vector_type(2))) float v2f;
typedef __attribute__((ext_vector_type(8))) float v8f;

#define NN 512
#define DD 512

// ---------------------------------------------------------------------------
// Kernel 1: X = h_e @ Wa^T ; Y = h_e @ Wb^T + W0_b   (f32 WMMA 16x16x4)
// One wave per 16x16 output tile. Tiles 0..1023 -> X, 1024..2047 -> Y.
// A layout (16x4 f32): lanes 0-15 hold M=0-15, VGPR0=K0,VGPR1=K1; lanes 16-31 K=2,3.
// B layout (4x16 f32): lane n holds column n; hi half-wave holds K=2,3.
// C/D layout (16x16 f32): VGPR r -> row r (lanes 0-15) / row 8+r (lanes 16-31).
// ---------------------------------------------------------------------------
__global__ __launch_bounds__(256) void pre_gemm_kernel(
    const float* __restrict__ h_e, const float* __restrict__ W0w,
    const float* __restrict__ W0b, float* __restrict__ X, float* __restrict__ Y) {
  const int wid  = threadIdx.x >> 5;
  const int lane = threadIdx.x & 31;
  const int lo = lane & 15;
  const int hi = lane >> 4;

  int t = blockIdx.x * 8 + wid;          // 0..2047 (wave-uniform)
  const int isY = (t >= 1024);
  if (isY) t -= 1024;
  const int ti = t >> 5;                 // i-tile  0..31
  const int th = t & 31;                 // h-tile  0..31
  const int i0 = ti * 16, h0 = th * 16;
  const int koff = isY ? DD : 0;         // Wb starts at column D of W0_w

  const float* arow = h_e + (size_t)(i0 + lo) * DD;                // A row (M = lo)
  const float* brow = W0w + (size_t)(h0 + lo) * (2 * DD) + koff;   // B col (N = lo)

  v8f acc = {};
  for (int k0 = 0; k0 < DD; k0 += 4) {
    v2f a, b;
    a.x = arow[k0 + hi * 2 + 0];
    a.y = arow[k0 + hi * 2 + 1];
    b.x = brow[k0 + hi * 2 + 0];
    b.y = brow[k0 + hi * 2 + 1];
    acc = __builtin_amdgcn_wmma_f32_16x16x4_f32(false, a, false, b,
                                                (short)0, acc, false, false);
  }

  const float bias = isY ? W0b[h0 + lo] : 0.0f;
  float* Out = isY ? Y : X;
#pragma unroll
  for (int r = 0; r < 8; ++r)
    Out[(size_t)(i0 + hi * 8 + r) * DD + h0 + lo] = acc[r] + bias;
}

// ---------------------------------------------------------------------------
// Kernel 2: per row i:  p[j] = sum_h relu(X[i,h] + Y[j,h]) * w[h];
//           a[i,:] = softmax(p).   (W1_b dropped: softmax is shift-invariant)
// x[h], w[h] live in 16 registers per lane; Y rows streamed coalesced.
// ---------------------------------------------------------------------------
__global__ __launch_bounds__(256) void pair_softmax_kernel(
    const float* __restrict__ X, const float* __restrict__ Y,
    const float* __restrict__ W1w, float* __restrict__ Amat) {
  __shared__ float p[NN];
  __shared__ float red[256];

  const int i = blockIdx.x;
  const int tid = threadIdx.x;
  const int wid = tid >> 5, lane = tid & 31;

  float xr[16], wr[16];
#pragma unroll
  for (int c = 0; c < 16; ++c) {
    xr[c] = X[(size_t)i * DD + c * 32 + lane];
    wr[c] = W1w[c * 32 + lane];
  }

  for (int j = wid; j < NN; j += 8) {
    const float* yrow = Y + (size_t)j * DD;
    float s = 0.0f;
#pragma unroll
    for (int c = 0; c < 16; ++c) {
      float v = xr[c] + yrow[c * 32 + lane];
      v = fmaxf(v, 0.0f);
      s = fmaf(v, wr[c], s);
    }
#pragma unroll
    for (int off = 16; off > 0; off >>= 1)
      s += __shfl_xor(s, off, 32);
    if (lane == 0) p[j] = s;
  }
  __syncthreads();

  // --- softmax over p[0..511] (each thread owns 2 entries) ---
  float m = fmaxf(p[tid], p[tid + 256]);
  red[tid] = m;
  __syncthreads();
  for (int st = 128; st > 0; st >>= 1) {
    if (tid < st) red[tid] = fmaxf(red[tid], red[tid + st]);
    __syncthreads();
  }
  m = red[0];
  __syncthreads();

  const float e0 = expf(p[tid] - m);
  const float e1 = expf(p[tid + 256] - m);
  red[tid] = e0 + e1;
  __syncthreads();
  for (int st = 128; st > 0; st >>= 1) {
    if (tid < st) red[tid] += red[tid + st];
    __syncthreads();
  }
  const float inv = 1.0f / red[0];

  Amat[(size_t)i * NN + tid]       = e0 * inv;
  Amat[(size_t)i * NN + tid + 256] = e1 * inv;
}

// ---------------------------------------------------------------------------
// Kernel 3: out = A @ h_e   (f32 WMMA 16x16x4), one wave per 16x16 tile.
// ---------------------------------------------------------------------------
__global__ __launch_bounds__(256) void out_gemm_kernel(
    const float* __restrict__ Amat, const float* __restrict__ h_e,
    float* __restrict__ out) {
  const int wid  = threadIdx.x >> 5;
  const int lane = threadIdx.x & 31;
  const int lo = lane & 15;
  const int hi = lane >> 4;

  const int t  = blockIdx.x * 8 + wid;   // 0..1023
  const int ti = t >> 5, td = t & 31;
  const int i0 = ti * 16, d0 = td * 16;

  const float* arow = Amat + (size_t)(i0 + lo) * NN;

  v8f acc = {};
  for (int k0 = 0; k0 < NN; k0 += 4) {
    v2f a, b;
    a.x = arow[k0 + hi * 2 + 0];
    a.y = arow[k0 + hi * 2 + 1];
    b.x = h_e[(size_t)(k0 + hi * 2 + 0) * DD + d0 + lo];
    b.y = h_e[(size_t)(k0 + hi * 2 + 1) * DD + d0 + lo];
    acc = __builtin_amdgcn_wmma_f32_16x16x4_f32(false, a, false, b,
                                                (short)0, acc, false, false);
  }
#pragma unroll
  for (int r = 0; r < 8; ++r)
    out[(size_t)(i0 + hi * 8 + r) * DD + d0 + lo] = acc[r];
}

// ---------------------------------------------------------------------------
extern "C" void kernel_launch(void* const* d_in, const int* in_sizes, int n_in,
                              void* d_out, int out_size, void* d_ws, size_t ws_size,
                              hipStream_t stream) {
  const float* h_e = (const float*)d_in[0];   // (512, 512)
  const float* W0w = (const float*)d_in[1];   // (512, 1024)
  const float* W0b = (const float*)d_in[2];   // (512,)
  const float* W1w = (const float*)d_in[3];   // (1, 512)
  // d_in[4] = W1_b : constant shift before softmax -> no effect, ignored.

  float* X    = (float*)d_ws;          // 512*512 f32  = 1 MB
  float* Y    = X + NN * DD;           // 1 MB
  float* Amat = Y + NN * DD;           // 1 MB
  float* out  = (float*)d_out;         // (512, 512) f32

  // 2048 wave-tiles / 8 waves per 256-thread block
  pre_gemm_kernel<<<256, 256, 0, stream>>>(h_e, W0w, W0b, X, Y);
  // one block per output row i
  pair_softmax_kernel<<<NN, 256, 0, stream>>>(X, Y, W1w, Amat);
  // 1024 wave-tiles / 8 waves per block
  out_gemm_kernel<<<128, 256, 0, stream>>>(Amat, h_e, out);
}